// HashEmbeddingBag_8169027797102
// MI455X (gfx1250) — compile-verified
//
#include <hip/hip_runtime.h>
#include <stdint.h>

// HashEmbeddingBag: out[b,d] = sum_{i in bag b} hashed_weight[ weight_idx[x[i]*64 + d] ]
//
// Strategy (MI455X / gfx1250):
//  * one wave32 per bag; lane t owns dims {2t, 2t+1}
//  * weight_idx rows (256B) are DMA'd global->LDS with the CDNA5 async engine
//    (global_load_async_to_lds_b64, ASYNCcnt, s_wait_asynccnt), 4-deep pipeline,
//    8 LDS slots per wave; consume stage unrolled 2x -> 4 outstanding random
//    gathers per wave + 2 independent accumulator pairs
//  * hashed_weight (16MB) is L2-resident (192MB L2); weight_idx streaming
//    (~210MB) sets the HBM floor (~10us @ 23.3TB/s)

#define WPB   8      // waves per block (wave32)
#define NBUF  8      // LDS row slots per wave
#define PIPE  4      // async prefetch depth (must be <= NBUF-2 for 2-wide consume)
#define DIM   64     // embedding dim

// ROCm 7.2's async builtin has a non-upstream signature; inline asm is the
// portable path. Operand order per VGLOBAL encoding: VDST (LDS byte-offset
// VGPR), VADDR (64-bit per-lane address pair), SADDR ('off' => GV mode).
__device__ __forceinline__ void async_row_b64(const int* gsrc, unsigned lds_off) {
  asm volatile("global_load_async_to_lds_b64 %0, %1, off"
               :: "v"(lds_off), "v"(gsrc)
               : "memory");
}

#if __has_builtin(__builtin_amdgcn_s_wait_asynccnt)
#define WAIT_ASYNCCNT(n) __builtin_amdgcn_s_wait_asynccnt(n)
#else
#define WAIT_ASYNCCNT(n) asm volatile("s_wait_asynccnt %0" :: "i"(n) : "memory")
#endif

#define MEMFENCE() asm volatile("" ::: "memory")

__global__ __launch_bounds__(32 * WPB, 1)
void hash_embedding_bag_kernel(const int* __restrict__ x,
                               const int* __restrict__ offsets,
                               const float* __restrict__ hashed_weight,
                               const int* __restrict__ weight_idx,
                               float* __restrict__ out,
                               int B, int N) {
  __shared__ __align__(16) int smem[WPB * NBUF * DIM];

  const int lane = threadIdx.x & 31;
  const int wave = threadIdx.x >> 5;
  const int bag  = blockIdx.x * WPB + wave;
  if (bag >= B) return;  // bag is wave-uniform -> whole wave exits together

  const int start = offsets[bag];
  const int end   = (bag + 1 < B) ? offsets[bag + 1] : N;
  const int cnt   = end - start;

  float2* outp = (float2*)(out + (size_t)bag * DIM + 2 * lane);
  if (cnt <= 0) { *outp = make_float2(0.f, 0.f); return; }

  int* buf = smem + wave * (NBUF * DIM);
  // low 32 bits of a generic shared pointer == LDS byte offset (aperture layout)
  const unsigned lds_base = (unsigned)(uintptr_t)buf + 8u * (unsigned)lane;

  // Preload this bag's token ids into lane registers; broadcast via v_readlane.
  int xa = 0, xb = 0;
  if (lane < cnt)      xa = x[start + lane];
  if (lane + 32 < cnt) xb = x[start + 32 + lane];

  auto xval = [&](int i) -> int {
    if (i < 32) return __builtin_amdgcn_readlane(xa, i);
    if (i < 64) return __builtin_amdgcn_readlane(xb, i - 32);
    return x[start + i];  // uniform address -> scalar load (rare: cnt > 64)
  };

  // Issue one async row fetch: 32 lanes x 8B = one 256B weight_idx row -> LDS slot.
  auto issue = [&](int j) {
    const int jj = (j < cnt) ? j : (cnt - 1);  // clamped tail keeps wait math exact
    const int xi = xval(jj);
    const int* gsrc = weight_idx + (size_t)xi * DIM + 2 * lane;
    const unsigned ldst = lds_base + (unsigned)((j & (NBUF - 1)) * DIM * 4);
    async_row_b64(gsrc, ldst);
  };

  for (int j = 0; j < PIPE; ++j) issue(j);   // issued through row PIPE-1

  float a0 = 0.f, a1 = 0.f, b0 = 0.f, b1 = 0.f;
  int i = 0;

  // Main loop: consume 2 rows per trip (4 in-flight gathers, 2 acc pairs).
  for (; i + 1 < cnt; i += 2) {
    // invariant: issued through i+PIPE-1; <=PIPE-2 younger than row i+1 =>
    // waiting ASYNCcnt<=PIPE-2 guarantees rows i and i+1 have landed (in-order).
    WAIT_ASYNCCNT(PIPE - 2);
    MEMFENCE();
    const int2 wA = *(const int2*)(buf + ((i)     & (NBUF - 1)) * DIM + 2 * lane);
    const int2 wB = *(const int2*)(buf + ((i + 1) & (NBUF - 1)) * DIM + 2 * lane);
    a0 += hashed_weight[wA.x];
    a1 += hashed_weight[wA.y];
    b0 += hashed_weight[wB.x];
    b1 += hashed_weight[wB.y];
    MEMFENCE();
    issue(i + PIPE);        // slots (i+4)&7,(i+5)&7: consumed 2 pairs ago -> safe
    issue(i + 1 + PIPE);
  }

  // Odd tail row.
  if (i < cnt) {
    WAIT_ASYNCCNT(PIPE - 1);
    MEMFENCE();
    const int2 w = *(const int2*)(buf + (i & (NBUF - 1)) * DIM + 2 * lane);
    a0 += hashed_weight[w.x];
    a1 += hashed_weight[w.y];
  }

  *outp = make_float2(a0 + b0, a1 + b1);
}

extern "C" void kernel_launch(void* const* d_in, const int* in_sizes, int n_in,
                              void* d_out, int out_size, void* d_ws, size_t ws_size,
                              hipStream_t stream) {
  const int*   x       = (const int*)d_in[0];    // [N]
  const int*   offsets = (const int*)d_in[1];    // [B]
  const float* hw      = (const float*)d_in[2];  // [HW]
  const int*   widx    = (const int*)d_in[3];    // [V, 64]
  float*       out     = (float*)d_out;          // [B, 64]

  const int N = in_sizes[0];
  const int B = in_sizes[1];

  const int blocks = (B + WPB - 1) / WPB;
  hipLaunchKernelGGL(hash_embedding_bag_kernel,
                     dim3(blocks), dim3(32 * WPB), 0, stream,
                     x, offsets, hw, widx, out, B, N);
}